// VectorField_58574763983264
// MI455X (gfx1250) — compile-verified
//
#include <hip/hip_runtime.h>

#define B_   2
#define N_   1024
#define D_   3
#define A_   64
#define R_   50
#define RPAD 64

typedef float v2f __attribute__((ext_vector_type(2)));
typedef float v8f __attribute__((ext_vector_type(8)));

// ---------------------------------------------------------------------------
// Kernel 1: prefactor = (emb[z] * time_rbf(t)) @ W   via V_WMMA_F32_16X16X4_F32
//   grid = (128 m-tiles, 4 r-tiles), block = 32 (one wave per 16x16 out tile)
//   P is stored padded: [B*N, RPAD]
// ---------------------------------------------------------------------------
__global__ __launch_bounds__(32)
void prefactor_wmma_kernel(const float* __restrict__ emb,
                           const int*   __restrict__ z,
                           const float* __restrict__ t,
                           const float* __restrict__ time_off,
                           const float* __restrict__ time_wid,
                           const float* __restrict__ W,
                           float* __restrict__ P) {
  const int mt   = blockIdx.x;      // row tile of BN (2048/16 = 128)
  const int rt   = blockIdx.y;      // col tile of RPAD (64/16 = 4)
  const int lane = threadIdx.x;     // 0..31
  const int g    = lane >> 4;       // lane group (0: lanes 0-15, 1: lanes 16-31)
  const int lr   = lane & 15;

  // A-fragment: this lane supplies row m = mt*16 + lr of X
  const int m  = mt * 16 + lr;
  const int b  = m >> 10;           // m / N_
  const int i  = m & (N_ - 1);      // m % N_
  const int zi = z[i];
  const float tb = t[b];

  // B-fragment: this lane supplies column r of Wpad (zero-padded 50 -> 64).
  // Clamp the index so the load is always in-bounds, select-zero afterwards:
  // keeps EXEC all-ones (no execz branches) as required around WMMA.
  const int  r      = rt * 16 + lr;
  const bool rvalid = (r < R_);
  const int  rc     = rvalid ? r : 0;

  v8f c = {0.f, 0.f, 0.f, 0.f, 0.f, 0.f, 0.f, 0.f};

  #pragma unroll
  for (int k0 = 0; k0 < A_; k0 += 4) {
    // 32-bit A layout: vgpr v, lane-group g -> K = k0 + v + 2*g
    const int ka = k0 + 2 * g;

    v2f a, bf;
    {
      const float o0 = time_off[ka],     w0 = time_wid[ka];
      const float o1 = time_off[ka + 1], w1 = time_wid[ka + 1];
      const float c0 = -0.5f * __builtin_amdgcn_rcpf(w0 * w0);  // v_rcp_f32
      const float c1 = -0.5f * __builtin_amdgcn_rcpf(w1 * w1);
      const float d0 = tb - o0, d1 = tb - o1;
      const float e0 = __expf(c0 * d0 * d0);                    // v_exp_f32
      const float e1 = __expf(c1 * d1 * d1);
      a.x = emb[zi * A_ + ka]     * e0;   // X[m, ka]
      a.y = emb[zi * A_ + ka + 1] * e1;   // X[m, ka+1]
    }
    // 32-bit B layout (K x N = 4 x 16): vgpr v, group g -> K = k0 + v + 2*g, N = lr
    const float w0v = W[ka * R_ + rc];
    const float w1v = W[(ka + 1) * R_ + rc];
    bf.x = rvalid ? w0v : 0.f;            // v_cndmask, no branch
    bf.y = rvalid ? w1v : 0.f;

    // 8 args: (neg_a, A, neg_b, B, c_mod, C, reuse_a, reuse_b)
    c = __builtin_amdgcn_wmma_f32_16x16x4_f32(false, a, false, bf,
                                              (short)0, c, false, false);
  }

  // C/D layout: vgpr v, group g -> M = v + 8*g, N = lr
  #pragma unroll
  for (int v = 0; v < 8; ++v) {
    const int row = mt * 16 + v + 8 * g;
    P[row * RPAD + rt * 16 + lr] = c[v];
  }
}

// ---------------------------------------------------------------------------
// Kernel 2: per (b,i) — pairwise distances, radial RBF, phi contraction,
//           weighted sum of r_ij, residual add.  Transcendental-bound
//   (~105M v_exp_f32 total).  grid = (N_, B_), block = 256.
// ---------------------------------------------------------------------------
__global__ __launch_bounds__(256)
void field_kernel(const float* __restrict__ pos,
                  const float* __restrict__ P,
                  const float* __restrict__ rad_off,
                  const float* __restrict__ rad_wid,
                  float* __restrict__ out) {
  const int i   = blockIdx.x;
  const int b   = blockIdx.y;
  const int tid = threadIdx.x;

  __shared__ float sx[N_], sy[N_], sz[N_];
  __shared__ float sp[R_], sc[R_], so[R_];
  __shared__ float red[3][256];

  const float* pb = pos + (size_t)b * N_ * D_;
  for (int j = tid; j < N_; j += 256) {
    sx[j] = pb[j * 3 + 0];
    sy[j] = pb[j * 3 + 1];
    sz[j] = pb[j * 3 + 2];
  }
  if (tid < R_) {
    sp[tid] = P[((size_t)b * N_ + i) * RPAD + tid];
    const float w = rad_wid[tid];
    sc[tid] = -0.5f * __builtin_amdgcn_rcpf(w * w);   // v_rcp_f32, no div expansion
    so[tid] = rad_off[tid];
  }
  __syncthreads();

  const float xi = sx[i], yi = sy[i], zi = sz[i];
  float ax = 0.f, ay = 0.f, az = 0.f;

  for (int j = tid; j < N_; j += 256) {
    if (j == i) continue;
    const float dx = xi - sx[j];
    const float dy = yi - sy[j];
    const float dz = zi - sz[j];
    const float d  = sqrtf(dx * dx + dy * dy + dz * dz + 1e-6f);
    float phi = 0.f;
    #pragma unroll
    for (int rr = 0; rr < R_; ++rr) {
      const float df = d - so[rr];
      phi += sp[rr] * __expf(sc[rr] * df * df);
    }
    ax += phi * dx;
    ay += phi * dy;
    az += phi * dz;
  }

  red[0][tid] = ax; red[1][tid] = ay; red[2][tid] = az;
  __syncthreads();
  #pragma unroll
  for (int s = 128; s > 0; s >>= 1) {
    if (tid < s) {
      red[0][tid] += red[0][tid + s];
      red[1][tid] += red[1][tid + s];
      red[2][tid] += red[2][tid + s];
    }
    __syncthreads();
  }
  if (tid == 0) {
    const size_t o = ((size_t)b * N_ + i) * 3;
    out[o + 0] = red[0][0] + pb[i * 3 + 0];
    out[o + 1] = red[1][0] + pb[i * 3 + 1];
    out[o + 2] = red[2][0] + pb[i * 3 + 2];
  }
}

// ---------------------------------------------------------------------------
extern "C" void kernel_launch(void* const* d_in, const int* in_sizes, int n_in,
                              void* d_out, int out_size, void* d_ws, size_t ws_size,
                              hipStream_t stream) {
  (void)in_sizes; (void)n_in; (void)out_size; (void)ws_size;
  const float* positions = (const float*)d_in[0];   // [B,N,D]
  const float* t         = (const float*)d_in[1];   // [B]
  const int*   z         = (const int*)  d_in[2];   // [N]
  const float* emb       = (const float*)d_in[3];   // [MAXZ,A]
  const float* W         = (const float*)d_in[4];   // [A,R]
  const float* rad_off   = (const float*)d_in[5];   // [R]
  const float* rad_wid   = (const float*)d_in[6];   // [R]
  const float* time_off  = (const float*)d_in[7];   // [A]
  const float* time_wid  = (const float*)d_in[8];   // [A]
  float* out = (float*)d_out;
  float* P   = (float*)d_ws;                        // [B*N, RPAD] = 512 KB

  dim3 g1(128, 4), b1(32);
  prefactor_wmma_kernel<<<g1, b1, 0, stream>>>(emb, z, t, time_off, time_wid, W, P);

  dim3 g2(N_, B_), b2(256);
  field_kernel<<<g2, b2, 0, stream>>>(positions, P, rad_off, rad_wid, out);
}